// BasisOrbitalBackflow_12429635355205
// MI455X (gfx1250) — compile-verified
//
#include <hip/hip_runtime.h>
#include <hip/hip_bf16.h>

// out[b,i,o] = (1/127) * sum_{k,l} (S[b,k]-chi[b,i,k]) * chi[b,i,l] * W[k*32+l, o]
// S[b,k] = sum_j chi[b,j,k].  B=8, N=128, NB=32, NORB=128.
// GEMM view: out = A @ W, A rows are rank-1 (u x v) built on the fly in registers.

typedef __attribute__((ext_vector_type(2))) float v2f;
typedef __attribute__((ext_vector_type(8))) float v8f;
typedef __attribute__((ext_vector_type(4))) int   v4i;

#define NB    32
#define NROW  128      // N (electrons) per batch
#define NORB  128
#define KTOT  1024     // NB*NB
#define KCH   256      // K rows of W staged in LDS per chunk
#define NCHUNK (KTOT / KCH)
#define CHI_STRIDE 33  // padded LDS row stride (bank-conflict free)

#if defined(__AMDGCN__) && \
    __has_builtin(__builtin_amdgcn_global_load_async_to_lds_b128) && \
    __has_builtin(__builtin_amdgcn_s_wait_asynccnt)
#define USE_ASYNC_LDS 1
typedef __attribute__((address_space(3))) v4i lds_v4i;
#else
#define USE_ASYNC_LDS 0
#endif

__global__ __launch_bounds__(256) void backflow_wmma_f32(
    const float* __restrict__ chi,   // (8,128,32)
    const float* __restrict__ W,     // (1024,128)
    float* __restrict__ out)         // (8,128,128)
{
    __shared__ float lds_chi[NROW * CHI_STRIDE];   // chi[b] padded        (16.9 KB)
    __shared__ float wtile[2][KCH * 16];           // double-buffered W    (32 KB)
    __shared__ float lds_part[8 * NB];
    __shared__ float lds_S[NB];

    const int t     = threadIdx.x;
    const int lane  = t & 31;
    const int w     = t >> 5;        // wave id 0..7 -> row stripe
    const int m     = lane & 15;     // M/N position within tile
    const int khalf = lane >> 4;     // 0: K+0/K+1, 1: K+2/K+3 (A,B frag halves)
    const int nbase = blockIdx.x * 16;
    const int b     = blockIdx.y;

    // Per-thread share of a W chunk: 4 x b128 (rr = p*64 + t/4 rows, q = t%4 float4 slab)
    const int q  = t & 3;
    const int r0 = t >> 2;

    // ---- stage chi[b] into LDS (padded) ----
    const float* chiB = chi + b * (NROW * NB);
    #pragma unroll
    for (int j = 0; j < 4; ++j) {
        float4 v = reinterpret_cast<const float4*>(chiB)[t + j * 256];
        int f = (t + j * 256) * 4;          // flat float index, row-aligned (32%4==0)
        int row = f >> 5, col = f & 31;
        float* p = &lds_chi[row * CHI_STRIDE + col];
        p[0] = v.x; p[1] = v.y; p[2] = v.z; p[3] = v.w;
    }

#if USE_ASYNC_LDS
    // ---- kick off async copy of W chunk 0 while we reduce column sums ----
    #pragma unroll
    for (int p = 0; p < 4; ++p) {
        int rr = p * 64 + r0;
        __builtin_amdgcn_global_load_async_to_lds_b128(
            (v4i*)(W + rr * NORB + nbase + q * 4),
            (lds_v4i*)&wtile[0][rr * 16 + q * 4], 0, 0);
    }
#endif
    __syncthreads();

    // ---- column sums S[k] over the 128 rows ----
    {
        int k = t & 31, g = t >> 5;
        float s = 0.f;
        #pragma unroll
        for (int r = 0; r < 16; ++r) s += lds_chi[(g * 16 + r) * CHI_STRIDE + k];
        lds_part[g * NB + k] = s;
    }
    __syncthreads();
    if (t < NB) {
        float s = 0.f;
        #pragma unroll
        for (int g = 0; g < 8; ++g) s += lds_part[g * NB + t];
        lds_S[t] = s;
    }

    const int   i      = w * 16 + m;         // electron row this lane feeds (A side)
    const float inv    = 1.0f / 127.0f;      // 1/(N-1)
    const float* chiRow = &lds_chi[i * CHI_STRIDE];

    v8f acc = {0.f, 0.f, 0.f, 0.f, 0.f, 0.f, 0.f, 0.f};

#if USE_ASYNC_LDS
    // wait for chunk 0 (per-wave ASYNCcnt), then block-wide visibility
    __builtin_amdgcn_s_wait_asynccnt(0);
    __syncthreads();   // also covers lds_S

    for (int kc = 0; kc < NCHUNK; ++kc) {
        const float* wt = wtile[kc & 1];
        // issue async copy of next chunk into the other buffer
        if (kc + 1 < NCHUNK) {
            #pragma unroll
            for (int p = 0; p < 4; ++p) {
                int rr = p * 64 + r0;
                __builtin_amdgcn_global_load_async_to_lds_b128(
                    (v4i*)(W + ((kc + 1) * KCH + rr) * NORB + nbase + q * 4),
                    (lds_v4i*)&wtile[(kc + 1) & 1][rr * 16 + q * 4], 0, 0);
            }
        }
#else
    for (int kc = 0; kc < NCHUNK; ++kc) {
        const float* wt = wtile[kc & 1];
        __syncthreads();                       // S ready / previous chunk consumed
        #pragma unroll
        for (int p = 0; p < 4; ++p) {
            int rr = p * 64 + r0;
            float4 v = *reinterpret_cast<const float4*>(W + (kc * KCH + rr) * NORB + nbase + q * 4);
            *reinterpret_cast<float4*>(&wtile[kc & 1][rr * 16 + q * 4]) = v;
        }
        __syncthreads();
#endif

        #pragma unroll
        for (int kk = 0; kk < 8; ++kk) {
            const int k = kc * 8 + kk;
            const float u_val = (lds_S[k] - chiRow[k]) * inv;   // u[i,k]
            #pragma unroll
            for (int lc = 0; lc < 8; ++lc) {
                // A fragment (16x4 f32): lane holds A[m, K0], A[m, K0+1]; K0 = lc*4 + 2*khalf
                const int lA = lc * 4 + khalf * 2;
                v2f a;
                a.x = u_val * chiRow[lA];
                a.y = u_val * chiRow[lA + 1];
                // B fragment (4x16 f32): lane holds W[K0, n], W[K0+1, n]; n = m
                const int rB = (kk * 32 + lA) * 16 + m;
                v2f bb;
                bb.x = wt[rB];
                bb.y = wt[rB + 16];
                acc = __builtin_amdgcn_wmma_f32_16x16x4_f32(
                    false, a, false, bb, (short)0, acc, false, false);
            }
        }

#if USE_ASYNC_LDS
        // next buffer loaded + all waves done reading current before it is reused
        if (kc + 1 < NCHUNK) {
            __builtin_amdgcn_s_wait_asynccnt(0);
            __syncthreads();
        }
#endif
    }

    // ---- D layout: VGPR j -> (M = j + 8*khalf, N = m) ----
    float* outB = out + (b * NROW) * NORB + nbase + m;
    #pragma unroll
    for (int j = 0; j < 8; ++j) {
        int row = w * 16 + j + khalf * 8;
        outB[row * NORB] = acc[j];
    }
}

extern "C" void kernel_launch(void* const* d_in, const int* in_sizes, int n_in,
                              void* d_out, int out_size, void* d_ws, size_t ws_size,
                              hipStream_t stream) {
    const float* chi = (const float*)d_in[0];            // 8*128*32
    const float* W   = (const float*)d_in[1];            // 1024*128
    float* out       = (float*)d_out;                    // 8*128*128
    (void)in_sizes; (void)n_in; (void)out_size; (void)d_ws; (void)ws_size;

    dim3 grid(NORB / 16, 8);   // (col tile, batch)
    dim3 block(256);           // 8 waves: one 16-row stripe each
    backflow_wmma_f32<<<grid, block, 0, stream>>>(chi, W, out);
}